// dynamic_reg_85323820302416
// MI455X (gfx1250) — compile-verified
//
#include <hip/hip_runtime.h>

// ---------------------------------------------------------------------------
// Graph WaveNet GCN block, fused for MI455X (gfx1250, wave32, WMMA).
//
//   out[b,o,w,l] = bias[o] + sum_{c'} W[o,c'] * h[b,c',w,l]
//   h = concat(x, A1^T x, (A1^2)^T x, A2^T x, (A2^2)^T x, A3^T x, (A3^2)^T x)
//
// Second-order diffusion folded via A^2 precompute; all 6 diffusions and the
// 1x1 conv run on V_WMMA_F32_16X16X4_F32, fused through LDS (h never hits
// memory). Diffusion GEMMs use two independent accumulator chains to break
// the WMMA RAW dependency; Y exchange buffer is double-buffered to halve
// barrier count; next channel-block of x is prefetched (global_prefetch_b8).
// ---------------------------------------------------------------------------

typedef float v2f __attribute__((ext_vector_type(2)));
typedef float v8f __attribute__((ext_vector_type(8)));

#define N_NODES 207
#define NPAD    208     // padded node dim (13 tiles of 16, K padded for WMMA)
#define VS      210     // LDS row stride in floats (even, gcd(210,64)=2)
#define CIN     32
#define COUT    32
#define LL      64
#define K7      224     // 7 * CIN
#define YSZ     2560    // one Y buffer: 256 columns x 10 (8 used, pad)

// LDS layout (float offsets)
#define WOFF  0                         // W[32][224]                  7168
#define ATOFF (WOFF + COUT * K7)        // AT[6][16][VS]              20160
#define XOFF  (ATOFF + 6 * 16 * VS)     // X[8][16][VS] (l-major)     26880
#define YOFF  (XOFF + 8 * 16 * VS)      // Y[2][256][10] (n-major)     5120
#define BOFF  (YOFF + 2 * YSZ)          // bias[32]
#define SMEM_FLOATS (BOFF + 32)         // 59360 floats = 237440 B (< 320 KB)

// ---------------------------------------------------------------------------
// Prep: mats[m] (208x208, row-major [w][v], zero padded) for m = 0..5:
//   m=2k   : A_{k+1}^T        (AT[w][v] = A[v][w])
//   m=2k+1 : (A_{k+1}^2)^T    (ATsq[w][v] = sum_u A[v][u] A[u][w])
// Matches W channel-block order [x, A1, A1^2, A2, A2^2, A3, A3^2].
// ---------------------------------------------------------------------------
__global__ void gcn_prep(const float* __restrict__ A1,
                         const float* __restrict__ A2,
                         const float* __restrict__ A3,
                         float* __restrict__ mats) {
  const int k = blockIdx.z;
  const float* A = (k == 0) ? A1 : ((k == 1) ? A2 : A3);
  const int w = blockIdx.y * 16 + threadIdx.y;
  const int v = blockIdx.x * 16 + threadIdx.x;
  if (w >= NPAD || v >= NPAD) return;

  float at = 0.0f, sq = 0.0f;
  if (w < N_NODES && v < N_NODES) {
    at = A[v * N_NODES + w];
    for (int u = 0; u < N_NODES; ++u)
      sq += A[v * N_NODES + u] * A[u * N_NODES + w];
  }
  mats[(2 * k)     * NPAD * NPAD + w * NPAD + v] = at;
  mats[(2 * k + 1) * NPAD * NPAD + w * NPAD + v] = sq;
}

// ---------------------------------------------------------------------------
// Fused diffusion + 1x1 conv. Grid: (L/16, 13, B). Block: 256 (8 waves).
// ---------------------------------------------------------------------------
__global__ __launch_bounds__(256)
void gcn_fused(const float* __restrict__ x,
               const float* __restrict__ mats,
               const float* __restrict__ Wg,
               const float* __restrict__ bg,
               float* __restrict__ out) {
  __shared__ float sm[SMEM_FLOATS];

  const int tid  = threadIdx.x;
  const int lane = tid & 31;
  const int wave = tid >> 5;          // 0..7 : channel-within-block / n-tile owner
  const int half = lane >> 4;         // lane 0-15 vs 16-31
  const int lm   = lane & 15;
  const int l0 = blockIdx.x * 16;
  const int w0 = blockIdx.y * 16;
  const int b  = blockIdx.z;

  // --- stage W (row-major [o][224]), bias, and the 6 A^T tile-row slabs ---
  for (int i = tid; i < COUT * K7; i += 256) sm[WOFF + i] = Wg[i];
  if (tid < COUT) sm[BOFF + tid] = bg[tid];
  for (int i = tid; i < 6 * 16 * NPAD; i += 256) {
    int m = i / (16 * NPAD);
    int r = (i / NPAD) & 15;
    int v = i % NPAD;
    sm[ATOFF + (m * 16 + r) * VS + v] = mats[(m * NPAD + (w0 + r)) * NPAD + v];
  }

  // Output accumulators: 2 o-tiles x 2 n-tiles (n = w*16+l columns) per wave.
  v8f oacc[2][2];
  #pragma unroll
  for (int a = 0; a < 2; ++a)
    #pragma unroll
    for (int j = 0; j < 2; ++j)
      oacc[a][j] = (v8f){0.f, 0.f, 0.f, 0.f, 0.f, 0.f, 0.f, 0.f};
  const int nt0 = 2 * wave;

  for (int cb = 0; cb < 4; ++cb) {           // 4 blocks of 8 input channels
    // stage X block transposed: sm X[cc][l][v] = x[b, 8cb+cc, v, l0+l]; v=207 -> 0
    // (safe without a leading barrier: previous chunk's consumers touch only Y/W)
    for (int q = tid; q < 8 * NPAD * 4; q += 256) {
      int cc = q / (NPAD * 4);
      int rv = q % (NPAD * 4);
      int v  = rv >> 2;
      int j4 = rv & 3;
      float4 f = make_float4(0.f, 0.f, 0.f, 0.f);
      if (v < N_NODES) {
        const float* src =
            &x[(((b * CIN + cb * 8 + cc) * N_NODES + v) * LL) + l0 + 4 * j4];
        f = *(const float4*)src;
        // warm next channel-block's lines (one prefetch per 64B segment)
        if (cb < 3 && j4 == 0)
          __builtin_prefetch(src + 8 * N_NODES * LL, 0, 0);
      }
      float* d = &sm[XOFF + (cc * 16 + 4 * j4) * VS + v];
      d[0] = f.x; d[VS] = f.y; d[2 * VS] = f.z; d[3 * VS] = f.w;
    }
    __syncthreads();

    for (int m = 0; m < 7; ++m) {
      const int ybase = YOFF + ((cb * 7 + m) & 1) * YSZ;   // double-buffered Y

      // ---- produce Y[c', 16x16 tile] for K-chunk c' = m*32 + cb*8 + wave ----
      if (m == 0) {
        // identity block: h = x at node rows w0..w0+15 (already staged)
        #pragma unroll
        for (int r = 0; r < 8; ++r) {
          int w = r + 8 * half;
          sm[ybase + (w * 16 + lm) * 10 + wave] =
              sm[XOFF + (wave * 16 + lm) * VS + (w0 + w)];
        }
      } else {
        // 16x16x208 GEMM: Y = AT_{m-1}[w-tile,:] * X[wave], two indep chains
        const float* pa = &sm[ATOFF + ((m - 1) * 16 + lm) * VS + 2 * half];
        const float* pb = &sm[XOFF  + (wave * 16 + lm) * VS + 2 * half];
        v8f acc0 = (v8f){0.f, 0.f, 0.f, 0.f, 0.f, 0.f, 0.f, 0.f};
        v8f acc1 = (v8f){0.f, 0.f, 0.f, 0.f, 0.f, 0.f, 0.f, 0.f};
        #pragma unroll 2
        for (int s = 0; s < 52; s += 2) {
          v2f a0 = *(const v2f*)(pa + 4 * s);       // A[M=lm, K=4s+2h+{0,1}]
          v2f b0 = *(const v2f*)(pb + 4 * s);       // B[K=4s+2h+{0,1}, N=lm]
          v2f a1 = *(const v2f*)(pa + 4 * s + 4);
          v2f b1 = *(const v2f*)(pb + 4 * s + 4);
          acc0 = __builtin_amdgcn_wmma_f32_16x16x4_f32(
              false, a0, false, b0, (short)0, acc0, false, false);
          acc1 = __builtin_amdgcn_wmma_f32_16x16x4_f32(
              false, a1, false, b1, (short)0, acc1, false, false);
        }
        #pragma unroll
        for (int r = 0; r < 8; ++r) {
          int w = r + 8 * half;
          sm[ybase + (w * 16 + lm) * 10 + wave] = acc0[r] + acc1[r];
        }
      }
      __syncthreads();

      // ---- consume: out[o, n] += W[o, chunk] * Y[chunk, n], K-chunk = 8 ----
      // No trailing barrier: next produce writes the other Y buffer, and no
      // wave passes the next barrier until all waves finished this consume.
      #pragma unroll
      for (int s = 0; s < 2; ++s) {
        int kidx = m * 32 + cb * 8 + 4 * s + 2 * half;     // even -> b64 aligned
        v2f a0 = *(const v2f*)&sm[WOFF + (0  + lm) * K7 + kidx];
        v2f a1 = *(const v2f*)&sm[WOFF + (16 + lm) * K7 + kidx];
        #pragma unroll
        for (int j = 0; j < 2; ++j) {
          int n = (nt0 + j) * 16 + lm;
          v2f bv = *(const v2f*)&sm[ybase + n * 10 + 4 * s + 2 * half];
          oacc[0][j] = __builtin_amdgcn_wmma_f32_16x16x4_f32(
              false, a0, false, bv, (short)0, oacc[0][j], false, false);
          oacc[1][j] = __builtin_amdgcn_wmma_f32_16x16x4_f32(
              false, a1, false, bv, (short)0, oacc[1][j], false, false);
        }
      }
    }
    __syncthreads();   // X block reuse across cb iterations
  }

  // ---- epilogue: bias + coalesced store; mask padded node row 207 ----
  #pragma unroll
  for (int j = 0; j < 2; ++j) {
    int w = nt0 + j;                 // n-tile == one node row, l = lm
    if (w0 + w < N_NODES) {
      #pragma unroll
      for (int ot = 0; ot < 2; ++ot) {
        #pragma unroll
        for (int r = 0; r < 8; ++r) {
          int o = ot * 16 + r + 8 * half;
          out[((b * COUT + o) * N_NODES + (w0 + w)) * LL + l0 + lm] =
              oacc[ot][j][r] + sm[BOFF + o];
        }
      }
    }
  }
}

extern "C" void kernel_launch(void* const* d_in, const int* in_sizes, int n_in,
                              void* d_out, int out_size, void* d_ws, size_t ws_size,
                              hipStream_t stream) {
  const float* x  = (const float*)d_in[0];
  const float* A1 = (const float*)d_in[1];
  const float* A2 = (const float*)d_in[2];
  const float* A3 = (const float*)d_in[3];
  const float* Wg = (const float*)d_in[4];
  const float* bg = (const float*)d_in[5];
  float* out  = (float*)d_out;
  float* mats = (float*)d_ws;   // 6 * 208 * 208 floats ~ 1.04 MB

  dim3 pb(16, 16, 1), pg(13, 13, 3);
  gcn_prep<<<pg, pb, 0, stream>>>(A1, A2, A3, mats);

  dim3 gb(256, 1, 1), gg(LL / 16, 13, 64);   // (l-tiles, w-tiles, batch)
  gcn_fused<<<gg, gb, 0, stream>>>(x, mats, Wg, bg, out);
}